// GAT_84696755077619
// MI455X (gfx1250) — compile-verified
//
#include <hip/hip_runtime.h>
#include <math.h>

typedef __attribute__((ext_vector_type(2))) float v2f;
typedef __attribute__((ext_vector_type(8))) float v8f;

__device__ inline void atomicMaxF32(float* addr, float val) {
  // monotonic mapping: signed-max for non-negative patterns, unsigned-min for negative
  int iv = __float_as_int(val);
  if (iv >= 0) atomicMax((int*)addr, iv);
  else atomicMin((unsigned int*)addr, (unsigned int)iv);
}

__device__ inline float leaky02(float t) { return t > 0.f ? t : 0.2f * t; }

// ---------------- GEMM: Out[N,256] = A[N,256] @ W[256,256] (fp32 WMMA) -------------
__global__ __launch_bounds__(256) void gemm256(const float* __restrict__ A,
                                               const float* __restrict__ W,
                                               float* __restrict__ Out, int N) {
  __shared__ float As[16 * 258];            // +2 pad: spreads LDS banks, keeps 8B align
  const int m0 = blockIdx.x * 16;
  const int t = threadIdx.x;
  // stage 16x256 A tile (coalesced global reads)
  for (int i = 0; i < 16; ++i) {
    int idx = t + i * 256;
    int r = idx >> 8, c = idx & 255;
    int gr = m0 + r; if (gr >= N) gr = N - 1;
    As[r * 258 + c] = A[gr * 256 + c];
  }
  __syncthreads();

  const int wave = t >> 5;                  // 8 waves
  const int lane = t & 31;
  const int half = lane >> 4;               // 0: K,K+1  1: K+2,K+3
  const int l16  = lane & 15;
  const int n0   = wave * 32;               // each wave: N-tiles n0 and n0+16

  v8f acc0 = {}; v8f acc1 = {};
  for (int k0 = 0; k0 < 256; k0 += 4) {
    const int ka = k0 + half * 2;
    v2f a;  a.x  = As[l16 * 258 + ka];          a.y  = As[l16 * 258 + ka + 1];
    v2f b0; b0.x = W[ka * 256 + n0 + l16];      b0.y = W[(ka + 1) * 256 + n0 + l16];
    v2f b1; b1.x = W[ka * 256 + n0 + 16 + l16]; b1.y = W[(ka + 1) * 256 + n0 + 16 + l16];
    acc0 = __builtin_amdgcn_wmma_f32_16x16x4_f32(false, a, false, b0, (short)0, acc0, false, false);
    acc1 = __builtin_amdgcn_wmma_f32_16x16x4_f32(false, a, false, b1, (short)0, acc1, false, false);
  }

  if (m0 + 16 <= N) {                       // uniform fast path: whole tile in range
    #pragma unroll
    for (int v = 0; v < 8; ++v) {
      int m = m0 + v + half * 8;
      Out[m * 256 + n0 + l16]      = acc0[v];
      Out[m * 256 + n0 + 16 + l16] = acc1[v];
    }
  } else {
    for (int v = 0; v < 8; ++v) {
      int m = m0 + v + half * 8;
      if (m < N) {
        Out[m * 256 + n0 + l16]      = acc0[v];
        Out[m * 256 + n0 + 16 + l16] = acc1[v];
      }
    }
  }
}

// ---------------- per-node attention logits: one wave per (node, head) -------------
__global__ __launch_bounds__(256) void alpha_kernel(const float* __restrict__ h,
    const float* __restrict__ a_s, const float* __restrict__ a_d,
    float* __restrict__ as_out, float* __restrict__ ad_out, int N) {
  int gw   = (blockIdx.x * 256 + threadIdx.x) >> 5;
  int lane = threadIdx.x & 31;
  int node = gw >> 2, head = gw & 3;
  if (node >= N) return;
  const float* hp = h + node * 256 + head * 64;
  const float* vs = a_s + head * 64;
  const float* vd = a_d + head * 64;
  float s = hp[lane] * vs[lane] + hp[lane + 32] * vs[lane + 32];
  float d = hp[lane] * vd[lane] + hp[lane + 32] * vd[lane + 32];
  for (int off = 16; off > 0; off >>= 1) {
    s += __shfl_down(s, off, 32);
    d += __shfl_down(d, off, 32);
  }
  if (lane == 0) { as_out[node * 4 + head] = s; ad_out[node * 4 + head] = d; }
}

// ---------------- zero accumulators / init segment max ----------------------------
__global__ __launch_bounds__(256) void init_kernel(float* __restrict__ agg,
    float* __restrict__ mx, float* __restrict__ dn, int N) {
  int i = blockIdx.x * 256 + threadIdx.x;
  if (i < N * 256) agg[i] = 0.f;
  if (i < N * 4) { mx[i] = -INFINITY; dn[i] = 0.f; }
}

// ---------------- pass 1: segment max over destinations ---------------------------
__global__ __launch_bounds__(256) void edge_max_kernel(const int* __restrict__ src,
    const int* __restrict__ dst, int E, int N,
    const float4* __restrict__ as_, const float4* __restrict__ ad_,
    float* __restrict__ mx) {
  int e = blockIdx.x * 256 + threadIdx.x;
  if (e >= E + N) return;
  int s, d;
  if (e < E) { s = src[e]; d = dst[e]; } else { s = d = e - E; }
  float4 a = as_[s];
  float4 b = ad_[d];
  atomicMaxF32(&mx[d * 4 + 0], leaky02(a.x + b.x));
  atomicMaxF32(&mx[d * 4 + 1], leaky02(a.y + b.y));
  atomicMaxF32(&mx[d * 4 + 2], leaky02(a.z + b.z));
  atomicMaxF32(&mx[d * 4 + 3], leaky02(a.w + b.w));
}

// ---------------- pass 2: segment sum of exp(e - m) -------------------------------
__global__ __launch_bounds__(256) void edge_sum_kernel(const int* __restrict__ src,
    const int* __restrict__ dst, int E, int N,
    const float4* __restrict__ as_, const float4* __restrict__ ad_,
    const float4* __restrict__ mx, float* __restrict__ dn) {
  int e = blockIdx.x * 256 + threadIdx.x;
  if (e >= E + N) return;
  int s, d;
  if (e < E) { s = src[e]; d = dst[e]; } else { s = d = e - E; }
  float4 a = as_[s];
  float4 b = ad_[d];
  float4 m = mx[d];
  atomicAdd(&dn[d * 4 + 0], expf(leaky02(a.x + b.x) - m.x));
  atomicAdd(&dn[d * 4 + 1], expf(leaky02(a.y + b.y) - m.y));
  atomicAdd(&dn[d * 4 + 2], expf(leaky02(a.z + b.z) - m.z));
  atomicAdd(&dn[d * 4 + 3], expf(leaky02(a.w + b.w) - m.w));
}

// ---------------- pass 3: weighted scatter-add (one wave per edge) ----------------
__global__ __launch_bounds__(256) void edge_agg_kernel(const int* __restrict__ src,
    const int* __restrict__ dst, int E, int N,
    const float4* __restrict__ as_, const float4* __restrict__ ad_,
    const float4* __restrict__ mx, const float4* __restrict__ dn,
    const float* __restrict__ h, float* __restrict__ agg) {
  int gw   = (blockIdx.x * 256 + threadIdx.x) >> 5;
  int lane = threadIdx.x & 31;
  if (gw >= E + N) return;
  int s, d;
  if (gw < E) { s = src[gw]; d = dst[gw]; } else { s = d = gw - E; }
  float4 a = as_[s];          // broadcast loads (same addr across wave)
  float4 b = ad_[d];
  float4 m = mx[d];
  float4 q = dn[d];
  float al[4];
  al[0] = expf(leaky02(a.x + b.x) - m.x) / q.x;
  al[1] = expf(leaky02(a.y + b.y) - m.y) / q.y;
  al[2] = expf(leaky02(a.z + b.z) - m.z) / q.z;
  al[3] = expf(leaky02(a.w + b.w) - m.w) / q.w;
  const float* hs = h + s * 256;
  float* op = agg + d * 256;
  #pragma unroll
  for (int hd = 0; hd < 4; ++hd) {
    int c = hd * 64 + lane;
    atomicAdd(&op[c],      hs[c]      * al[hd]);   // fire-and-forget f32 atomics
    atomicAdd(&op[c + 32], hs[c + 32] * al[hd]);
  }
}

// ---------------- bias + exact GELU ------------------------------------------------
__global__ __launch_bounds__(256) void bias_gelu_kernel(const float* __restrict__ in,
    const float* __restrict__ b, float* __restrict__ out, int total) {
  int i = blockIdx.x * 256 + threadIdx.x;
  if (i >= total) return;
  float y = in[i] + b[i & 255];
  out[i] = 0.5f * y * (1.0f + erff(y * 0.70710678118654752f));
}

extern "C" void kernel_launch(void* const* d_in, const int* in_sizes, int n_in,
                              void* d_out, int out_size, void* d_ws, size_t ws_size,
                              hipStream_t stream) {
  const float* x   = (const float*)d_in[0];
  const int*   ei  = (const int*)d_in[1];
  // d_in[2] edge_attr unused (edge_dim=None)
  const float* W0  = (const float*)d_in[3];
  const float* as0 = (const float*)d_in[4];
  const float* ad0 = (const float*)d_in[5];
  const float* b0  = (const float*)d_in[6];
  const float* W1  = (const float*)d_in[7];
  const float* as1 = (const float*)d_in[8];
  const float* ad1 = (const float*)d_in[9];
  const float* b1  = (const float*)d_in[10];
  float* out = (float*)d_out;

  const int N  = in_sizes[0] / 256;
  const int E  = in_sizes[1] / 2;
  const int ET = E + N;
  const int* src = ei;
  const int* dst = ei + E;

  float* ws  = (float*)d_ws;
  float* h   = ws;                          // N*256
  float* agg = h   + (size_t)N * 256;       // N*256 (also reused as activation)
  float* asb = agg + (size_t)N * 256;       // N*4  (16B aligned)
  float* adb = asb + (size_t)N * 4;         // N*4
  float* mx  = adb + (size_t)N * 4;         // N*4
  float* dn  = mx  + (size_t)N * 4;         // N*4

  const float4* asb4 = (const float4*)asb;
  const float4* adb4 = (const float4*)adb;
  const float4* mx4  = (const float4*)mx;
  const float4* dn4  = (const float4*)dn;

  dim3 blk(256);
  const int gemmGrid  = (N + 15) / 16;
  const int alphaGrid = (N * 4 * 32 + 255) / 256;
  const int initGrid  = (N * 256 + 255) / 256;
  const int edgeGrid  = (ET + 255) / 256;
  const int aggGrid   = (ET * 32 + 255) / 256;

  // ---- layer 1 ----
  gemm256<<<gemmGrid, blk, 0, stream>>>(x, W0, h, N);
  alpha_kernel<<<alphaGrid, blk, 0, stream>>>(h, as0, ad0, asb, adb, N);
  init_kernel<<<initGrid, blk, 0, stream>>>(agg, mx, dn, N);
  edge_max_kernel<<<edgeGrid, blk, 0, stream>>>(src, dst, E, N, asb4, adb4, mx);
  edge_sum_kernel<<<edgeGrid, blk, 0, stream>>>(src, dst, E, N, asb4, adb4, mx4, dn);
  edge_agg_kernel<<<aggGrid, blk, 0, stream>>>(src, dst, E, N, asb4, adb4, mx4, dn4, h, agg);
  bias_gelu_kernel<<<initGrid, blk, 0, stream>>>(agg, b0, agg, N * 256);   // in-place act

  // ---- layer 2 ----
  gemm256<<<gemmGrid, blk, 0, stream>>>(agg, W1, h, N);
  alpha_kernel<<<alphaGrid, blk, 0, stream>>>(h, as1, ad1, asb, adb, N);
  init_kernel<<<initGrid, blk, 0, stream>>>(agg, mx, dn, N);   // act is dead after gemm
  edge_max_kernel<<<edgeGrid, blk, 0, stream>>>(src, dst, E, N, asb4, adb4, mx);
  edge_sum_kernel<<<edgeGrid, blk, 0, stream>>>(src, dst, E, N, asb4, adb4, mx4, dn);
  edge_agg_kernel<<<aggGrid, blk, 0, stream>>>(src, dst, E, N, asb4, adb4, mx4, dn4, h, agg);
  bias_gelu_kernel<<<initGrid, blk, 0, stream>>>(agg, b1, out, N * 256);
}